// YoloV1Loss_6605659701993
// MI455X (gfx1250) — compile-verified
//
#include <hip/hip_runtime.h>

// MI455X (gfx1250) YOLOv1 loss: memory-bound streaming reduction.
// ~385 MB total reads / 23.3 TB/s  =>  ~16.5 us floor. Strategy:
//   - coalesced non-temporal b128 global loads (read-once stream, skip L2
//     retention: 385 MB > 192 MB L2, zero reuse) -> LDS staging
//   - per-lane loss math on LDS (30-dword stride: bank-conflict-free in wave32)
//   - wave32 reduction via V_WMMA_F32_16X16X4_F32 (B = ones) + shfl_xor(16)
//   - deterministic two-pass block-partial reduction (no float atomics)

typedef float v2f __attribute__((ext_vector_type(2)));
typedef float v4f __attribute__((ext_vector_type(4)));
typedef float v8f __attribute__((ext_vector_type(8)));

#define NB1    1024   // blocks in pass 1 (= number of partials)
#define TPB    256    // threads per block (8 wave32 waves)
#define CELL_F 30     // floats per grid cell

__global__ void __launch_bounds__(TPB)
yolo_partial_kernel(const float* __restrict__ P, const float* __restrict__ T,
                    float* __restrict__ partial, int nCells, int nChunks)
{
    __shared__ float sp[TPB * CELL_F];      // 30 KB predict tile
    __shared__ float st[TPB * CELL_F];      // 30 KB target tile
    __shared__ float wsum[TPB / 32];        // one partial per wave

    const int tid = threadIdx.x;
    float acc = 0.0f;

    for (int chunk = blockIdx.x; chunk < nChunks; chunk += NB1) {
        const int cellBase = chunk * TPB;
        int cellsHere = nCells - cellBase;
        if (cellsHere > TPB) cellsHere = TPB;
        const int nFloats = cellsHere * CELL_F;

        const float* gp = P + (long long)cellBase * CELL_F;
        const float* gt = T + (long long)cellBase * CELL_F;

        if (nFloats == TPB * CELL_F) {
            // full chunk: 1920 float4 per tensor, coalesced b128 NT loads
            const v4f* gp4 = (const v4f*)gp;
            const v4f* gt4 = (const v4f*)gt;
            v4f* sp4 = (v4f*)sp;
            v4f* st4 = (v4f*)st;
            #pragma unroll 2
            for (int i = tid; i < (TPB * CELL_F) / 4; i += TPB) {
                sp4[i] = __builtin_nontemporal_load(gp4 + i);
                st4[i] = __builtin_nontemporal_load(gt4 + i);
            }
        } else {
            for (int i = tid; i < nFloats; i += TPB) {
                sp[i] = __builtin_nontemporal_load(gp + i);
                st[i] = __builtin_nontemporal_load(gt + i);
            }
        }
        __syncthreads();

        if (tid < cellsHere) {
            const float* pc = sp + tid * CELL_F;   // stride 30 dwords: conflict-free
            const float* tc = st + tid * CELL_F;

            const float t4    = tc[4];
            const float coord = (t4 > 0.0f)  ? 1.0f : 0.0f;
            const float noobj = (t4 == 0.0f) ? 1.0f : 0.0f;

            // target box 0 -> xyxy
            const float tww = tc[2], thh = tc[3];
            const float thx = 0.5f * tww, thy = 0.5f * thh;
            const float tx1 = tc[0] - thx, ty1 = tc[1] - thy;
            const float tx2 = tc[0] + thx, ty2 = tc[1] + thy;
            const float a2  = tww * thh;

            float iou0, iou1;
            #pragma unroll
            for (int k = 0; k < 2; ++k) {
                const int b5 = k * 5;
                const float pw  = pc[b5 + 2], ph = pc[b5 + 3];
                const float phx = 0.5f * pw, phy = 0.5f * ph;
                const float px1 = pc[b5] - phx, py1 = pc[b5 + 1] - phy;
                const float px2 = pc[b5] + phx, py2 = pc[b5 + 1] + phy;
                float iw = fminf(px2, tx2) - fmaxf(px1, tx1);
                float ih = fminf(py2, ty2) - fmaxf(py1, ty1);
                iw = fmaxf(iw, 0.0f);
                ih = fmaxf(ih, 0.0f);
                const float inter = iw * ih;
                const float a1    = pw * ph;
                const float iou   = inter / (a1 + a2 - inter);
                if (k == 0) iou0 = iou; else iou1 = iou;
            }

            // argmax with first-max tie rule (matches jnp.argmax)
            const int   roff   = (iou0 >= iou1) ? 0 : 5;
            const float maxiou = fmaxf(iou0, iou1);    // detached in reference
            const float* pr = pc + roff;
            const float* tr = tc + roff;
            const float pn4 = pc[9 - roff];            // non-responsible conf

            const float dx  = pr[0] - tr[0];
            const float dy  = pr[1] - tr[1];
            const float dsw = sqrtf(pr[2]) - sqrtf(tr[2]);   // inputs >= 0.05: safe
            const float dsh = sqrtf(pr[3]) - sqrtf(tr[3]);
            const float loc = dx * dx + dy * dy + dsw * dsw + dsh * dsh;

            const float dcon    = pr[4] - maxiou;
            const float contain = dcon * dcon;

            const float d4 = pc[4] - tc[4];
            const float d9 = pc[9] - tc[9];
            const float noobj_l = d4 * d4 + d9 * d9;

            float cls = 0.0f;
            #pragma unroll
            for (int cI = 10; cI < 30; ++cI) {
                const float d = pc[cI] - tc[cI];
                cls += d * d;
            }

            acc += coord * (5.0f * loc + 2.0f * contain + pn4 * pn4 + cls)
                 + 0.5f * noobj * noobj_l;
        }
        __syncthreads();
    }

    // ---- wave32 reduction via V_WMMA_F32_16X16X4_F32 ----
    // A (16x4): VGPR0 = acc -> A[m][0]=acc[m], A[m][2]=acc[m+16]; VGPR1 = 0.
    // B (4x16) = ones  =>  D[m][n] = acc[m] + acc[m+16] for every n.
    // Per-lane sum of the 8 D VGPRs = half-sum; + shfl_xor(16) = full 32-lane sum.
    v2f a; a[0] = acc;  a[1] = 0.0f;
    v2f b; b[0] = 1.0f; b[1] = 1.0f;
    v8f c = {};
    c = __builtin_amdgcn_wmma_f32_16x16x4_f32(
            /*neg_a=*/false, a, /*neg_b=*/false, b,
            /*c_mod=*/(short)0, c, /*reuse_a=*/false, /*reuse_b=*/false);
    float t8 = c[0] + c[1] + c[2] + c[3] + c[4] + c[5] + c[6] + c[7];
    const float wtot = t8 + __shfl_xor(t8, 16, 32);

    if ((tid & 31) == 0) wsum[tid >> 5] = wtot;
    __syncthreads();
    if (tid == 0) {
        float s = 0.0f;
        #pragma unroll
        for (int w = 0; w < TPB / 32; ++w) s += wsum[w];
        partial[blockIdx.x] = s;   // written unconditionally every call
    }
}

__global__ void __launch_bounds__(TPB)
yolo_reduce_kernel(const float* __restrict__ partial, float* __restrict__ out, float invN)
{
    __shared__ float s[TPB];
    const int tid = threadIdx.x;
    float v = 0.0f;
    #pragma unroll
    for (int i = tid; i < NB1; i += TPB) v += partial[i];
    s[tid] = v;
    __syncthreads();
    #pragma unroll
    for (int off = TPB / 2; off > 0; off >>= 1) {
        if (tid < off) s[tid] += s[tid + off];
        __syncthreads();
    }
    if (tid == 0) out[0] = s[0] * invN;   // deterministic fixed-order sum
}

extern "C" void kernel_launch(void* const* d_in, const int* in_sizes, int n_in,
                              void* d_out, int out_size, void* d_ws, size_t ws_size,
                              hipStream_t stream)
{
    const float* P = (const float*)d_in[0];   // predict [B,7,7,30] fp32
    const float* T = (const float*)d_in[1];   // target  [B,7,7,30] fp32
    float* partial = (float*)d_ws;            // NB1 floats of scratch

    const int nElems  = in_sizes[0];
    const int nCells  = nElems / CELL_F;          // B*49
    const int nChunks = (nCells + TPB - 1) / TPB; // 6272 for B=32768
    const int batch   = nCells / 49;
    const float invN  = 1.0f / (float)batch;

    yolo_partial_kernel<<<NB1, TPB, 0, stream>>>(P, T, partial, nCells, nChunks);
    yolo_reduce_kernel<<<1, TPB, 0, stream>>>(partial, (float*)d_out, invN);
}